// GraphAttentionLayer_36627481100951
// MI455X (gfx1250) — compile-verified
//
#include <hip/hip_runtime.h>
#include <hip/hip_bf16.h>

// ---------------------------------------------------------------------------
// GAT layer, sparse formulation for MI455X (gfx1250, wave32).
//   h = x @ W            -> v_wmma_f32_16x16x4_f32, A tile staged via TDM
//                           (tensor_load_to_lds with pad to 264-float rows)
//   s_src/s_dst          -> small GEMV
//   edge mask            -> 8192x8192 bitmask (exact duplicate-edge collapse)
//   softmax + aggregate  -> per-row block, bitmask scan, gather of h rows
// ---------------------------------------------------------------------------

typedef __attribute__((ext_vector_type(2))) float    v2f;
typedef __attribute__((ext_vector_type(8))) float    v8f;
typedef __attribute__((ext_vector_type(4))) unsigned u32x4;
typedef __attribute__((ext_vector_type(8))) int      i32x8;
typedef __attribute__((ext_vector_type(4))) int      i32x4;

#define N_NODES 8192
#define F_IN    256
#define F_OUT   64
#define MASK_WPR 256   /* 8192 bits / 32 = 256 u32 words per row */

#if __has_builtin(__builtin_amdgcn_tensor_load_to_lds)
#define HAVE_TDM 1
#else
#define HAVE_TDM 0
#endif

// ---------------------------------------------------------------------------
// Kernel 1: h = x @ W via V_WMMA_F32_16X16X4_F32.
// grid = 512 blocks (M tiles of 16), block = 128 threads = 4 waves.
// Wave w computes output columns [16w, 16w+16).
// A-tile (16x256 f32) staged in LDS by the Tensor Data Mover; the TDM pad
// feature (interval=256 DWORDs, amount=8 DWORDs) produces the 264-float row
// stride that spreads LDS banks.
// ---------------------------------------------------------------------------
__global__ __launch_bounds__(128) void gat_h_wmma(const float* __restrict__ x,
                                                  const float* __restrict__ W,
                                                  float* __restrict__ h) {
    __shared__ float As[16 * 264];          // 16 rows, stride 264 floats
    const int mt   = blockIdx.x;
    const int tid  = threadIdx.x;
    const int wave = tid >> 5;              // 0..3 -> N tile
    const int lane = tid & 31;
    const int m0   = mt * 16;

#if HAVE_TDM
    if (wave == 0) {
        // ---- Tensor DMA descriptor: 16x256 f32 tile, row-major, LDS-padded.
        const unsigned lds_base =
            (unsigned)(size_t)(void*)&As[0];          // addr[31:0] == LDS offset
        const unsigned long long ga =
            (unsigned long long)(const void*)(x + (size_t)m0 * F_IN);

        u32x4 g0;
        g0[0] = 1u;                                   // count=1, user mode
        g0[1] = lds_base;                             // lds_addr
        g0[2] = (unsigned)(ga & 0xFFFFFFFFull);       // global_addr[31:0]
        g0[3] = (unsigned)((ga >> 32) & 0x01FFFFFFull)// global_addr[56:32]
              | (2u << 30);                           // type=2 ("image")

        i32x8 g1;
        g1[0] = (int)((2u << 16)      // data_size = 4 bytes
                    | (1u << 20)      // pad_enable
                    | (7u << 22)      // pad_interval = 256 DWORDs (one row)
                    | (7u << 25));    // pad_amount   = 8 DWORDs
        g1[1] = (int)(256u << 16);    // tensor_dim0 = 256 (low 16 bits)
        g1[2] = (int)(8192u << 16);   // tensor_dim1 = 8192 (low 16 bits)
        g1[3] = (int)(256u << 16);    // tile_dim0 = 256
        g1[4] = 16;                   // tile_dim1 = 16, tile_dim2 = 0
        g1[5] = 256;                  // tensor_dim0_stride = 256
        g1[6] = 0;
        g1[7] = 0;

        i32x4 gz4 = {};
#if __clang_major__ >= 23
        i32x8 gz8 = {};
        __builtin_amdgcn_tensor_load_to_lds(g0, g1, gz4, gz4, gz8, 0);
#else
        __builtin_amdgcn_tensor_load_to_lds(g0, g1, gz4, gz4, 0);
#endif
        __builtin_amdgcn_s_wait_tensorcnt(0);
    }
    __syncthreads();
#else
    // Fallback staging: 16 rows x 64 float4 = 1024 float4, 8 per thread.
    const float4* xv = (const float4*)(x + (size_t)m0 * F_IN);
    for (int i = tid; i < 1024; i += 128) {
        const int r  = i >> 6;
        const int c4 = i & 63;
        float4 v = xv[r * 64 + c4];
        float* dst = &As[r * 264 + c4 * 4];
        dst[0] = v.x; dst[1] = v.y; dst[2] = v.z; dst[3] = v.w;
    }
    __syncthreads();
#endif

    const int lm  = lane & 15;              // row (A) / col (B,C,D) within tile
    const int hi  = lane >> 4;              // 0: K={0,1}, 1: K={2,3}
    const int col = wave * 16 + lm;

    // Pull this wave's W panel toward the caches (global_prefetch_b8).
    __builtin_prefetch(W + col, 0, 1);

    v8f acc = {};
    #pragma unroll 8
    for (int k = 0; k < F_IN; k += 4) {
        const int ko = k + hi * 2;
        v2f a;
        a.x = As[lm * 264 + ko];
        a.y = As[lm * 264 + ko + 1];
        v2f b;
        b.x = W[(ko)     * F_OUT + col];
        b.y = W[(ko + 1) * F_OUT + col];
        acc = __builtin_amdgcn_wmma_f32_16x16x4_f32(
            /*neg_a=*/false, a, /*neg_b=*/false, b,
            /*c_mod=*/(short)0, acc, /*reuse_a=*/false, /*reuse_b=*/false);
    }

    // D layout: VGPR r holds row (r + 8*hi), col = lane%16.
    #pragma unroll
    for (int r = 0; r < 8; ++r) {
        const int m = m0 + r + hi * 8;
        h[(size_t)m * F_OUT + col] = acc[r];
    }
}

// ---------------------------------------------------------------------------
// Kernel 2: s_src[m] = h[m,:] . a[:64], s_dst[m] = h[m,:] . a[64:]
// ---------------------------------------------------------------------------
__global__ void gat_scores(const float* __restrict__ h,
                           const float* __restrict__ a,
                           float* __restrict__ ssrc,
                           float* __restrict__ sdst) {
    const int m = blockIdx.x * blockDim.x + threadIdx.x;
    if (m < N_NODES) {
        float s1 = 0.f, s2 = 0.f;
        #pragma unroll 8
        for (int c = 0; c < F_OUT; ++c) {
            const float v = h[(size_t)m * F_OUT + c];
            s1 += v * a[c];
            s2 += v * a[F_OUT + c];
        }
        ssrc[m] = s1;
        sdst[m] = s2;
    }
}

// ---------------------------------------------------------------------------
// Kernel 3: column mean of h (reference semantics for edge-less rows:
// softmax over all-NEG row is uniform -> output = mean of h).
// ---------------------------------------------------------------------------
__global__ __launch_bounds__(64) void gat_meanh(const float* __restrict__ h,
                                                float* __restrict__ meanh) {
    const int c = threadIdx.x;
    float s = 0.f;
    for (int m = 0; m < N_NODES; ++m) s += h[(size_t)m * F_OUT + c];
    meanh[c] = s * (1.0f / (float)N_NODES);
}

// ---------------------------------------------------------------------------
// Kernel 4: clear bitmask.
// ---------------------------------------------------------------------------
__global__ void gat_clear(unsigned* __restrict__ mask, int nwords) {
    int i = blockIdx.x * blockDim.x + threadIdx.x;
    const int stride = gridDim.x * blockDim.x;
    for (; i < nwords; i += stride) mask[i] = 0u;
}

// ---------------------------------------------------------------------------
// Kernel 5: scatter edges into the bitmask (exact dedup, matches mask.set(1)).
// edge_index is int64 in the reference.
// ---------------------------------------------------------------------------
__global__ void gat_scatter(const long long* __restrict__ ei,
                            unsigned* __restrict__ mask, int E) {
    const int e = blockIdx.x * blockDim.x + threadIdx.x;
    if (e < E) {
        const int i = (int)ei[e];
        const int j = (int)ei[(size_t)E + e];
        atomicOr(&mask[(size_t)i * MASK_WPR + (j >> 5)], 1u << (j & 31));
    }
}

// ---------------------------------------------------------------------------
// Kernel 6: per-row masked softmax + aggregation.
// grid = 8192 blocks (one row), block = 64 threads (2 waves).
// ---------------------------------------------------------------------------
__global__ __launch_bounds__(64) void gat_row(const float* __restrict__ h,
                                              const float* __restrict__ ssrc,
                                              const float* __restrict__ sdst,
                                              const unsigned* __restrict__ mask,
                                              const float* __restrict__ meanh,
                                              float* __restrict__ out) {
    __shared__ unsigned mrow[MASK_WPR];
    __shared__ float red[64];

    const int i = blockIdx.x;
    const int t = threadIdx.x;
    const unsigned* mw = mask + (size_t)i * MASK_WPR;

    for (int w = t; w < MASK_WPR; w += 64) mrow[w] = mw[w];
    __syncthreads();

    const float si = ssrc[i];

    // Pass 1: row max of leaky_relu scores over set bits.
    float mx = -3.0e38f;
    for (int w = t; w < MASK_WPR; w += 64) {
        unsigned bits = mrow[w];
        while (bits) {
            const int b = __builtin_ctz(bits);
            bits &= bits - 1u;
            const int j = w * 32 + b;
            float v = si + sdst[j];
            v = v > 0.f ? v : 0.2f * v;
            mx = fmaxf(mx, v);
        }
    }
    red[t] = mx;
    __syncthreads();
    for (int s = 32; s > 0; s >>= 1) {
        if (t < s) red[t] = fmaxf(red[t], red[t + s]);
        __syncthreads();
    }
    const float rowmax = red[0];
    __syncthreads();

    // Pass 2: weighted gather; thread t = output column t.
    if (rowmax > -1.0e38f) {
        float accv = 0.f, sum = 0.f;
        for (int w = 0; w < MASK_WPR; ++w) {
            unsigned bits = mrow[w];
            while (bits) {
                const int b = __builtin_ctz(bits);
                bits &= bits - 1u;
                const int j = w * 32 + b;
                float v = si + sdst[j];
                v = v > 0.f ? v : 0.2f * v;
                const float wgt = __expf(v - rowmax);
                sum  += wgt;
                accv += wgt * h[(size_t)j * F_OUT + t];
            }
        }
        out[(size_t)i * F_OUT + t] = accv / sum;
    } else {
        // No edges: reference softmax degenerates to uniform 1/N.
        out[(size_t)i * F_OUT + t] = meanh[t];
    }
}

// ---------------------------------------------------------------------------
// Host-side launch.  Inputs: x[8192*256] f32, edge_index[2*E] i64,
// W[256*64] f32, a[128] f32.  Output: [8192*64] f32.
// ---------------------------------------------------------------------------
extern "C" void kernel_launch(void* const* d_in, const int* in_sizes, int n_in,
                              void* d_out, int out_size, void* d_ws, size_t ws_size,
                              hipStream_t stream) {
    const float*     x  = (const float*)d_in[0];
    const long long* ei = (const long long*)d_in[1];
    const float*     W  = (const float*)d_in[2];
    const float*     a  = (const float*)d_in[3];
    float*           out = (float*)d_out;
    const int E = in_sizes[1] / 2;

    char* ws = (char*)d_ws;
    float* h     = (float*)ws;  ws += (size_t)N_NODES * F_OUT * sizeof(float); // 2 MB
    float* ssrc  = (float*)ws;  ws += (size_t)N_NODES * sizeof(float);
    float* sdst  = (float*)ws;  ws += (size_t)N_NODES * sizeof(float);
    float* meanh = (float*)ws;  ws += 64 * sizeof(float);
    unsigned* mask = (unsigned*)ws;                                           // 8 MB

    gat_h_wmma <<<N_NODES / 16, 128, 0, stream>>>(x, W, h);
    gat_scores <<<N_NODES / 256, 256, 0, stream>>>(h, a, ssrc, sdst);
    gat_meanh  <<<1, 64, 0, stream>>>(h, meanh);
    gat_clear  <<<2048, 256, 0, stream>>>(mask, N_NODES * MASK_WPR);
    gat_scatter<<<(E + 255) / 256, 256, 0, stream>>>(ei, mask, E);
    gat_row    <<<N_NODES, 64, 0, stream>>>(h, ssrc, sdst, mask, meanh, out);
}